// Memory_41472204210735
// MI455X (gfx1250) — compile-verified
//
#include <hip/hip_runtime.h>

// Memory-update operator for MI455X (gfx1250, wave32, WMMA).
// bs=8, m=4096, K=256, V=3.
// Heavy part: S = softmax(k @ m_k^T) reduced to row stats (pass1) and
// column max (pass2) without materializing S. fp32 WMMA (16x16x4) keeps
// reference numerics for the discontinuous threshold/argsort stages.
// B tiles are double-buffered through the CDNA5 async global->LDS path
// (GLOBAL_LOAD_ASYNC_TO_LDS_B128 + s_wait_asynccnt) to overlap tile loads
// with the WMMA inner loop.

#define BS    8
#define MDIM  4096
#define KDIM  256
#define VDIM  3
#define THRESH (0.35f * 100.0f / 4096.0f)

typedef __attribute__((ext_vector_type(2))) float v2f;
typedef __attribute__((ext_vector_type(8))) float v8f;

#define ROWS_PER_BLOCK 64
#define APAD 4
#define ASTRIDE (KDIM + APAD)   // 260 floats: bank-conflict-free (260 % 64 == 4)

// --- CDNA5 async global->LDS copy (16B per lane), tracked by ASYNCcnt -------
__device__ __forceinline__ void async_b128(unsigned lds_off, const float* g) {
  asm volatile("global_load_async_to_lds_b128 %0, %1, off"
               :: "v"(lds_off), "v"(g) : "memory");
}
__device__ __forceinline__ void async_wait0() {
  asm volatile("s_wait_asynccnt 0x0" ::: "memory");
}
__device__ __forceinline__ unsigned lds_off_of(const void* p) {
  return (unsigned)(unsigned long long)p;   // low 32 bits of generic ptr = LDS offset
}

// ---------------------------------------------------------------------------
// Pass 1: per-row online softmax stats of S = k @ m_k^T : rowmax[h], Z[h]
// ---------------------------------------------------------------------------
__launch_bounds__(128)
__global__ void pass1_rowstats(const float* __restrict__ kmat,
                               const float* __restrict__ mk,
                               float* __restrict__ rowmax,
                               float* __restrict__ rowZ) {
  __shared__ float As[ROWS_PER_BLOCK * ASTRIDE];  // 64 x 256 k-stripe
  __shared__ float Bs[2][16 * ASTRIDE];           // double-buffered m_k tile
  const int b    = blockIdx.y;
  const int row0 = blockIdx.x * ROWS_PER_BLOCK;
  const int tid  = threadIdx.x;
  const int wave = tid >> 5;
  const int lane = tid & 31;
  const int mrow = lane & 15;           // M (for A) / N (for B) within tile
  const int khalf = (lane >> 4) << 1;   // K pair: lanes 0-15 -> K0,K1; 16-31 -> K2,K3

  // Async-stage the 64x256 A stripe (32 x b128 per thread).
  for (int x = tid; x < ROWS_PER_BLOCK * (KDIM / 4); x += 128) {
    const int r = x >> 6, c4 = x & 63;
    async_b128(lds_off_of(&As[r * ASTRIDE + c4 * 4]),
               &kmat[((size_t)b * MDIM + row0 + r) * KDIM + c4 * 4]);
  }
  // Async-stage B tile 0 into buffer 0.
  for (int x = tid; x < 16 * (KDIM / 4); x += 128) {
    const int c = x >> 6, c4 = x & 63;
    async_b128(lds_off_of(&Bs[0][c * ASTRIDE + c4 * 4]),
               &mk[((size_t)b * MDIM + c) * KDIM + c4 * 4]);
  }
  async_wait0();
  __syncthreads();

  float run_max[8], run_z[8];
#pragma unroll
  for (int r = 0; r < 8; ++r) { run_max[r] = -__builtin_huge_valf(); run_z[r] = 0.0f; }

  const int arow = wave * 16 + mrow;
  const int NT = MDIM / 16;

  for (int ct = 0; ct < NT; ++ct) {
    const int cur = ct & 1;
    // Kick off the next tile into the other buffer (overlaps the WMMAs).
    if (ct + 1 < NT) {
      const int nxt = (ct + 1) & 1;
      for (int x = tid; x < 16 * (KDIM / 4); x += 128) {
        const int c = x >> 6, c4 = x & 63;
        async_b128(lds_off_of(&Bs[nxt][c * ASTRIDE + c4 * 4]),
                   &mk[((size_t)b * MDIM + (ct + 1) * 16 + c) * KDIM + c4 * 4]);
      }
    }

    v8f acc = {0.f, 0.f, 0.f, 0.f, 0.f, 0.f, 0.f, 0.f};
#pragma unroll 8
    for (int kk = 0; kk < KDIM / 4; ++kk) {
      const v2f a  = *(const v2f*)&As[arow * ASTRIDE + kk * 4 + khalf];
      const v2f bb = *(const v2f*)&Bs[cur][mrow * ASTRIDE + kk * 4 + khalf];
      acc = __builtin_amdgcn_wmma_f32_16x16x4_f32(false, a, false, bb,
                                                  (short)0, acc, false, false);
    }
    // Online softmax update. C/D layout: VGPR r = row r (lanes 0-15) / r+8 (16-31).
#pragma unroll
    for (int r = 0; r < 8; ++r) {
      const float sv = acc[r];
      float tmax = sv;
      tmax = fmaxf(tmax, __shfl_xor(tmax, 1, 32));
      tmax = fmaxf(tmax, __shfl_xor(tmax, 2, 32));
      tmax = fmaxf(tmax, __shfl_xor(tmax, 4, 32));
      tmax = fmaxf(tmax, __shfl_xor(tmax, 8, 32));
      const float nm = fmaxf(run_max[r], tmax);
      float e = expf(sv - nm);
      e += __shfl_xor(e, 1, 32);
      e += __shfl_xor(e, 2, 32);
      e += __shfl_xor(e, 4, 32);
      e += __shfl_xor(e, 8, 32);
      run_z[r]   = run_z[r] * expf(run_max[r] - nm) + e;
      run_max[r] = nm;
    }

    async_wait0();
    __syncthreads();
  }

  if (mrow == 0) {
    const int base = b * MDIM + row0 + wave * 16 + (lane >> 4) * 8;
#pragma unroll
    for (int r = 0; r < 8; ++r) {
      rowmax[base + r] = run_max[r];
      rowZ[base + r]   = run_z[r];
    }
  }
}

// ---------------------------------------------------------------------------
// Pass 2: column max of softmax(S): colmax[m] = max_h exp(s_hm - rowmax_h)/Z_h
// (atomicMax on positive-float bit patterns; colmax pre-zeroed)
// ---------------------------------------------------------------------------
__launch_bounds__(128)
__global__ void pass2_colmax(const float* __restrict__ kmat,
                             const float* __restrict__ mk,
                             const float* __restrict__ rowmax,
                             const float* __restrict__ rowZ,
                             unsigned* __restrict__ colmax) {
  __shared__ float As[ROWS_PER_BLOCK * ASTRIDE];
  __shared__ float Bs[2][16 * ASTRIDE];
  const int b    = blockIdx.y;
  const int row0 = blockIdx.x * ROWS_PER_BLOCK;
  const int tid  = threadIdx.x;
  const int wave = tid >> 5;
  const int lane = tid & 31;
  const int mrow = lane & 15;
  const int khalf = (lane >> 4) << 1;

  for (int x = tid; x < ROWS_PER_BLOCK * (KDIM / 4); x += 128) {
    const int r = x >> 6, c4 = x & 63;
    async_b128(lds_off_of(&As[r * ASTRIDE + c4 * 4]),
               &kmat[((size_t)b * MDIM + row0 + r) * KDIM + c4 * 4]);
  }
  for (int x = tid; x < 16 * (KDIM / 4); x += 128) {
    const int c = x >> 6, c4 = x & 63;
    async_b128(lds_off_of(&Bs[0][c * ASTRIDE + c4 * 4]),
               &mk[((size_t)b * MDIM + c) * KDIM + c4 * 4]);
  }

  float rm[8], zi[8];
  const int rowbase = row0 + wave * 16 + (lane >> 4) * 8;  // rows for my half
#pragma unroll
  for (int r = 0; r < 8; ++r) {
    rm[r] = rowmax[b * MDIM + rowbase + r];
    zi[r] = 1.0f / rowZ[b * MDIM + rowbase + r];
  }

  async_wait0();
  __syncthreads();

  const int arow = wave * 16 + mrow;
  const int NT = MDIM / 16;

  for (int ct = 0; ct < NT; ++ct) {
    const int cur = ct & 1;
    if (ct + 1 < NT) {
      const int nxt = (ct + 1) & 1;
      for (int x = tid; x < 16 * (KDIM / 4); x += 128) {
        const int c = x >> 6, c4 = x & 63;
        async_b128(lds_off_of(&Bs[nxt][c * ASTRIDE + c4 * 4]),
                   &mk[((size_t)b * MDIM + (ct + 1) * 16 + c) * KDIM + c4 * 4]);
      }
    }

    v8f acc = {0.f, 0.f, 0.f, 0.f, 0.f, 0.f, 0.f, 0.f};
#pragma unroll 8
    for (int kk = 0; kk < KDIM / 4; ++kk) {
      const v2f a  = *(const v2f*)&As[arow * ASTRIDE + kk * 4 + khalf];
      const v2f bb = *(const v2f*)&Bs[cur][mrow * ASTRIDE + kk * 4 + khalf];
      acc = __builtin_amdgcn_wmma_f32_16x16x4_f32(false, a, false, bb,
                                                  (short)0, acc, false, false);
    }
    float vmax = 0.0f;
#pragma unroll
    for (int r = 0; r < 8; ++r)
      vmax = fmaxf(vmax, expf(acc[r] - rm[r]) * zi[r]);
    // lane n (rows 0..7 block) and lane n+16 (rows 8..15) hold the same column
    vmax = fmaxf(vmax, __shfl_xor(vmax, 16, 32));
    if (lane < 16)
      atomicMax(&colmax[b * MDIM + ct * 16 + lane], __float_as_uint(vmax));

    async_wait0();
    __syncthreads();
  }
}

// ---------------------------------------------------------------------------
// Stable argsort of 4096 keys per batch (bitonic on (key, idx) pairs).
// mode 0: key = keys[i]               (m_u ascending  -> idx_u)
// mode 1: key = -(1/keys[i])          (keys = rowZ    -> idx_d)
// mode 2: key = (1/keys[i] < TH)?0:1  (keys = rowZ, argsort(!wv) -> perm)
// ---------------------------------------------------------------------------
__launch_bounds__(1024)
__global__ void argsort_kernel(const float* __restrict__ keys_in, int mode,
                               int* __restrict__ idx_out) {
  __shared__ float skey[MDIM];
  __shared__ int   sidx[MDIM];
  const int b = blockIdx.x;
  for (int i = threadIdx.x; i < MDIM; i += blockDim.x) {
    const float kv = keys_in[b * MDIM + i];
    float key;
    if (mode == 0)      key = kv;
    else if (mode == 1) key = -(1.0f / kv);
    else                key = ((1.0f / kv) < THRESH) ? 0.0f : 1.0f;
    skey[i] = key;
    sidx[i] = i;
  }
  __syncthreads();
  for (int k2 = 2; k2 <= MDIM; k2 <<= 1) {
    for (int j = k2 >> 1; j > 0; j >>= 1) {
      for (int t = threadIdx.x; t < MDIM / 2; t += blockDim.x) {
        const int i   = ((t & ~(j - 1)) << 1) | (t & (j - 1));
        const int ixj = i | j;
        const bool up = ((i & k2) == 0);
        const float ka = skey[i], kb = skey[ixj];
        const int   ia = sidx[i], ib = sidx[ixj];
        const bool agtb = (ka > kb) || (ka == kb && ia > ib);  // lexicographic => stable
        if (agtb == up) {
          skey[i] = kb; skey[ixj] = ka;
          sidx[i] = ib; sidx[ixj] = ia;
        }
      }
      __syncthreads();
    }
  }
  for (int i = threadIdx.x; i < MDIM; i += blockDim.x)
    idx_out[b * MDIM + i] = sidx[i];
}

// ---------------------------------------------------------------------------
// Finalize: one wave per output row j.
// keep = wv[perm[j]]; keep: gather k/v via idx_d[perm[j]], m_u=1
//                    else : gather m_k/m_v/m_u via idx_u[j], m_u=.99*mu+colmax[j]
// ---------------------------------------------------------------------------
__launch_bounds__(256)
__global__ void finalize_kernel(const float* __restrict__ kmat,
                                const float* __restrict__ vmat,
                                const float* __restrict__ mkin,
                                const float* __restrict__ mvin,
                                const float* __restrict__ mu,
                                const int* __restrict__ idx_u,
                                const int* __restrict__ idx_d,
                                const int* __restrict__ perm,
                                const float* __restrict__ rowZ,
                                const unsigned* __restrict__ colmax,
                                float* __restrict__ out_mk,
                                float* __restrict__ out_mv,
                                float* __restrict__ out_mu) {
  const int gw   = blockIdx.x * 8 + (threadIdx.x >> 5);
  const int b    = gw >> 12;
  const int j    = gw & 4095;
  const int lane = threadIdx.x & 31;

  const int  p    = perm[b * MDIM + j];
  const bool keep = (1.0f / rowZ[b * MDIM + p]) < THRESH;

  const float *srck, *srcv;
  if (keep) {
    const int src = idx_d[b * MDIM + p];
    srck = &kmat[((size_t)b * MDIM + src) * KDIM];
    srcv = &vmat[((size_t)b * MDIM + src) * VDIM];
  } else {
    const int iu = idx_u[b * MDIM + j];
    srck = &mkin[((size_t)b * MDIM + iu) * KDIM];
    srcv = &mvin[((size_t)b * MDIM + iu) * VDIM];
  }
  const float4 x0 = *(const float4*)&srck[lane * 8];
  const float4 x1 = *(const float4*)&srck[lane * 8 + 4];
  float* dk = &out_mk[((size_t)b * MDIM + j) * KDIM];
  *(float4*)&dk[lane * 8]     = x0;
  *(float4*)&dk[lane * 8 + 4] = x1;

  if (lane < VDIM)
    out_mv[((size_t)b * MDIM + j) * VDIM + lane] = srcv[lane];

  if (lane == 31) {
    float val;
    if (keep) {
      val = 1.0f;
    } else {
      const int iu = idx_u[b * MDIM + j];
      val = 0.99f * mu[b * MDIM + iu] + __uint_as_float(colmax[b * MDIM + j]);
    }
    out_mu[b * MDIM + j] = val;
  }
}

// ---------------------------------------------------------------------------
extern "C" void kernel_launch(void* const* d_in, const int* in_sizes, int n_in,
                              void* d_out, int out_size, void* d_ws, size_t ws_size,
                              hipStream_t stream) {
  const float* kmat = (const float*)d_in[0];  // (8,4096,256)
  const float* vmat = (const float*)d_in[1];  // (8,4096,3)
  const float* mkin = (const float*)d_in[2];  // (8,4096,256)
  const float* mvin = (const float*)d_in[3];  // (8,4096,3)
  const float* mu   = (const float*)d_in[4];  // (8,4096)

  const int NM = BS * MDIM;                   // 32768
  float*    ws     = (float*)d_ws;
  float*    rowmax = ws;                      // NM floats
  float*    rowZ   = ws + NM;                 // NM floats
  unsigned* colmax = (unsigned*)(ws + 2 * NM);// NM u32 (float bits)
  int*      idx_u  = (int*)(ws + 3 * NM);     // NM ints
  int*      idx_d  = (int*)(ws + 4 * NM);     // NM ints
  int*      perm   = (int*)(ws + 5 * NM);     // NM ints

  float* out_mk = (float*)d_out;
  float* out_mv = out_mk + (size_t)BS * MDIM * KDIM;
  float* out_mu = out_mv + (size_t)BS * MDIM * VDIM;

  hipMemsetAsync(colmax, 0, (size_t)NM * sizeof(unsigned), stream);

  const dim3 ggrid(MDIM / ROWS_PER_BLOCK, BS);  // (64, 8)
  pass1_rowstats<<<ggrid, 128, 0, stream>>>(kmat, mkin, rowmax, rowZ);
  pass2_colmax  <<<ggrid, 128, 0, stream>>>(kmat, mkin, rowmax, rowZ, colmax);

  argsort_kernel<<<BS, 1024, 0, stream>>>(mu,   0, idx_u);
  argsort_kernel<<<BS, 1024, 0, stream>>>(rowZ, 1, idx_d);
  argsort_kernel<<<BS, 1024, 0, stream>>>(rowZ, 2, perm);

  finalize_kernel<<<NM / 8, 256, 0, stream>>>(kmat, vmat, mkin, mvin, mu,
                                              idx_u, idx_d, perm, rowZ, colmax,
                                              out_mk, out_mv, out_mu);
}